// ARBLoss_79439715106888
// MI455X (gfx1250) — compile-verified
//
#include <hip/hip_runtime.h>
#include <math.h>

// ARBLoss for MI455X (gfx1250, wave32).
// loss = -(1/n) * sum_i [ log(out[i,y_i]) + log(cnt[y_i]) - log(rowsum_i) - log(n) ]
// Row sums computed as GEMV-by-ones using V_WMMA_F32_16X16X4_F32.

typedef __attribute__((ext_vector_type(2))) float v2f;
typedef __attribute__((ext_vector_type(8))) float v8f;

#define N_ROWS 8192
#define N_COLS 32000
#define TILE_M 16
#define WAVES  8
#define BLOCK  (WAVES * 32)
#define COLS_PER_WAVE (N_COLS / WAVES)   // 4000, divisible by 8
#define NBLOCKS (N_ROWS / TILE_M)        // 512

__global__ void arb_zero_ws(int* __restrict__ counts, float* __restrict__ partials) {
  int i = blockIdx.x * blockDim.x + threadIdx.x;
  if (i < N_COLS) counts[i] = 0;
  if (i < NBLOCKS) partials[i] = 0.0f;
}

__global__ void arb_hist(const int* __restrict__ y, int* __restrict__ counts) {
  int i = blockIdx.x * blockDim.x + threadIdx.x;
  if (i < N_ROWS) atomicAdd(&counts[y[i]], 1);
}

__global__ __launch_bounds__(BLOCK)
void arb_rowsum_loss(const float* __restrict__ output,
                     const int*   __restrict__ y,
                     const int*   __restrict__ counts,
                     float*       __restrict__ partials) {
  __shared__ float lds_rowsum[WAVES * TILE_M];
  __shared__ float lds_term[TILE_M];

  const int tid  = threadIdx.x;
  const int wave = tid >> 5;     // wave32
  const int lane = tid & 31;
  const int m    = lane & 15;    // A-matrix row M = lane%16
  const int half = lane >> 4;    // lanes 16-31 hold K=2,3
  const int r0   = blockIdx.x * TILE_M;

  // Per-lane streaming pointer into this wave's column slice of row (r0+m).
  const float* p = output
                 + (size_t)(r0 + m) * N_COLS
                 + (size_t)(wave * COLS_PER_WAVE)
                 + 2 * half;

  v2f ones; ones[0] = 1.0f; ones[1] = 1.0f;   // B = 4x16 all-ones
  v8f acc0 = {};
  v8f acc1 = {};

  // D[m,n] = acc[m,n] + sum_k A[m,k] : 8 columns of 16 rows per iteration.
  #pragma unroll 2
  for (int it = 0; it < COLS_PER_WAVE / 8; ++it) {
    v2f a0 = *(const v2f*)(p);       // cols [c, c+4)  (this lane's K pair)
    v2f a1 = *(const v2f*)(p + 4);   // cols [c+4, c+8)
    acc0 = __builtin_amdgcn_wmma_f32_16x16x4_f32(
        false, a0, false, ones, (short)0, acc0, false, false);
    acc1 = __builtin_amdgcn_wmma_f32_16x16x4_f32(
        false, a1, false, ones, (short)0, acc1, false, false);
    p += 8;
  }
  v8f acc = acc0 + acc1;

  // C/D layout: lane 0 VGPR v = D[v, 0] (rows 0-7), lane 16 VGPR v = D[v+8, 0].
  if (lane == 0) {
    #pragma unroll
    for (int i = 0; i < 8; ++i) lds_rowsum[wave * TILE_M + i] = acc[i];
  } else if (lane == 16) {
    #pragma unroll
    for (int i = 0; i < 8; ++i) lds_rowsum[wave * TILE_M + 8 + i] = acc[i];
  }
  __syncthreads();

  // 16 threads: combine wave partial row sums, gather true-class prob, log terms.
  if (tid < TILE_M) {
    float rs = 0.0f;
    #pragma unroll
    for (int w = 0; w < WAVES; ++w) rs += lds_rowsum[w * TILE_M + tid];
    const int   row  = r0 + tid;
    const int   cls  = y[row];
    const float t    = output[(size_t)row * N_COLS + cls];
    const float wcnt = (float)counts[cls];
    lds_term[tid] = logf(t) + logf(wcnt) - logf(rs) - logf((float)N_ROWS);
  }
  __syncthreads();

  if (tid == 0) {
    float s = 0.0f;
    #pragma unroll
    for (int i = 0; i < TILE_M; ++i) s += lds_term[i];
    partials[blockIdx.x] = s;
  }
}

__global__ __launch_bounds__(NBLOCKS)
void arb_finalize(const float* __restrict__ partials, float* __restrict__ out) {
  __shared__ float lds[NBLOCKS];
  const int tid = threadIdx.x;
  lds[tid] = partials[tid];
  __syncthreads();
  for (int s = NBLOCKS / 2; s > 0; s >>= 1) {
    if (tid < s) lds[tid] += lds[tid + s];
    __syncthreads();
  }
  if (tid == 0) out[0] = -lds[0] / (float)N_ROWS;
}

extern "C" void kernel_launch(void* const* d_in, const int* in_sizes, int n_in,
                              void* d_out, int out_size, void* d_ws, size_t ws_size,
                              hipStream_t stream) {
  const float* output = (const float*)d_in[0];   // [8192, 32000] f32
  const int*   y      = (const int*)d_in[1];     // [8192] int
  float*       out    = (float*)d_out;           // scalar loss

  int*   counts   = (int*)d_ws;                                  // 32000 ints
  float* partials = (float*)((char*)d_ws + N_COLS * sizeof(int)); // 512 floats

  arb_zero_ws<<<(N_COLS + 255) / 256, 256, 0, stream>>>(counts, partials);
  arb_hist<<<(N_ROWS + 255) / 256, 256, 0, stream>>>(y, counts);
  arb_rowsum_loss<<<NBLOCKS, BLOCK, 0, stream>>>(output, y, counts, partials);
  arb_finalize<<<1, NBLOCKS, 0, stream>>>(partials, out);
}